// AdditiveAttention_8426725834780
// MI455X (gfx1250) — compile-verified
//
#include <hip/hip_runtime.h>
#include <hip/hip_bf16.h>
#include <math.h>

// AdditiveAttention, MI455X (gfx1250, wave32).
// Algebraic collapse: energies[b,n] = relu(K[b,n]·(Wk@Wh) + c_b),
//   c_b = Q[b]·(Wq@Wh) + (bk+bq)·Wh + bh.
// HBM-bound: stream K (128MB) once for energies, V (128MB) once for output.
// K·w uses V_WMMA_F32_16X16X4_F32 with w broadcast across the N dimension.

typedef __attribute__((ext_vector_type(2))) float v2f;
typedef __attribute__((ext_vector_type(8))) float v8f;

#define BQ 32
#define NKK 2048
#define DKK 512
#define DVV 512
#define HD  512

// workspace layout (floats)
#define WS_W   0                      // [512]  Wk@Wh
#define WS_U   512                    // [512]  Wq@Wh
#define WS_C   1024                   // [32]   per-batch additive constant
#define WS_E   1056                   // [32*2048] relu'd energies
#define WS_MAX (WS_E + BQ*NKK)        // [32]
#define WS_SUM (WS_MAX + BQ)          // [32]
#define WS_PART (WS_SUM + BQ)         // [32*16*4*128] partial V sums

__device__ inline float waveReduceSum(float v) {
#pragma unroll
    for (int off = 16; off > 0; off >>= 1) v += __shfl_down(v, off, 32);
    return v;
}

// ---- Kernel 1: w = Wk@Wh, u = Wq@Wh (1024 waves, one dot each) ----
__global__ void k_wu(const float* __restrict__ Wk, const float* __restrict__ Wq,
                     const float* __restrict__ Wh, float* __restrict__ w,
                     float* __restrict__ u) {
    int t = threadIdx.x, lane = t & 31;
    int idx = blockIdx.x * 8 + (t >> 5);          // 0..1023
    const float* row = (idx < 512 ? Wk : Wq) + (size_t)(idx & 511) * HD;
    float s = 0.f;
    for (int h = lane; h < HD; h += 32) s += row[h] * Wh[h];
    s = waveReduceSum(s);
    if (lane == 0) { if (idx < 512) w[idx] = s; else u[idx - 512] = s; }
}

// ---- Kernel 2: c[b] = Q[b]·u + (bk+bq)·Wh + bh (one wave per batch) ----
__global__ void k_ctx(const float* __restrict__ Q, const float* __restrict__ u,
                      const float* __restrict__ bk, const float* __restrict__ bq,
                      const float* __restrict__ Wh, const float* __restrict__ bh,
                      float* __restrict__ c) {
    int b = blockIdx.x, lane = threadIdx.x;
    float s = 0.f;
    for (int h = lane; h < HD; h += 32)
        s += Q[(size_t)b * HD + h] * u[h] + (bk[h] + bq[h]) * Wh[h];
    s = waveReduceSum(s);
    if (lane == 0) c[b] = s + bh[0];
}

// ---- Kernel 3: energies via WMMA f32 16x16x4, one wave per 16-row K tile ----
// A fragment: lane m(0..15)=row m; VGPR0/1 hold K cols {0,1} (lanes 0-15) or
// {2,3} (lanes 16-31) -> contiguous float2 per lane. B = w chunk replicated
// across all 16 N columns, so every D column equals the desired dot product.
__global__ void k_energy(const float* __restrict__ K, const float* __restrict__ w,
                         const float* __restrict__ c, float* __restrict__ e) {
    __shared__ __align__(16) float wlds[DKK];
    int t = threadIdx.x;
    for (int i = t; i < DKK; i += 256) wlds[i] = w[i];
    __syncthreads();

    int lane = t & 31;
    int tile = blockIdx.x * 8 + (t >> 5);         // 0..4095 (exact)
    int b = tile >> 7;                            // 128 tiles per batch
    int n0 = (tile & 127) << 4;
    const float* Kb = K + ((size_t)b * NKK + n0) * DKK;

    int m    = lane & 15;
    int kOff = (lane >> 4) << 1;                  // 0 or 2
    const float* aPtr = Kb + (size_t)m * DKK + kOff;
    const float* bPtr = wlds + kOff;

    v8f acc = {0.f, 0.f, 0.f, 0.f, 0.f, 0.f, 0.f, 0.f};
#pragma unroll 8
    for (int cc = 0; cc < DKK; cc += 4) {
        v2f a  = *(const v2f*)(aPtr + cc);        // global_load_b64
        v2f bb = *(const v2f*)(bPtr + cc);        // ds_load_b64
        acc = __builtin_amdgcn_wmma_f32_16x16x4_f32(
            false, a, false, bb, (short)0, acc, false, false);
    }

    float cb = c[b];
    // D layout: VGPR j -> (M=j, N=lane) lanes 0-15; (M=8+j, N=lane-16) for 16-31.
    // All N columns are equal, so lanes 0-7 carry rows 0-7 (j = lane), lanes
    // 16-23 carry rows 8-15. Branch-free cndmask select tree + one store.
    int j = lane & 7;
    float s01 = (j & 1) ? acc[1] : acc[0];
    float s23 = (j & 1) ? acc[3] : acc[2];
    float s45 = (j & 1) ? acc[5] : acc[4];
    float s67 = (j & 1) ? acc[7] : acc[6];
    float s03 = (j & 2) ? s23 : s01;
    float s47 = (j & 2) ? s67 : s45;
    float val = (j & 4) ? s47 : s03;
    if ((lane & 15) < 8) {
        int row = j + ((lane >> 4) << 3);
        e[(size_t)b * NKK + n0 + row] = fmaxf(val + cb, 0.0f);
    }
}

// ---- Kernel 4: per-batch softmax stats (max + sum of exp) ----
__global__ void k_stats(const float* __restrict__ e, float* __restrict__ mx,
                        float* __restrict__ sm) {
    __shared__ float red[256];
    int b = blockIdx.x, t = threadIdx.x;
    const float* eb = e + (size_t)b * NKK;
    float m = 0.f;                                // relu output >= 0
    for (int n = t; n < NKK; n += 256) m = fmaxf(m, eb[n]);
    red[t] = m; __syncthreads();
    for (int s = 128; s > 0; s >>= 1) { if (t < s) red[t] = fmaxf(red[t], red[t + s]); __syncthreads(); }
    float bm = red[0]; __syncthreads();
    float sum = 0.f;
    for (int n = t; n < NKK; n += 256) sum += __expf(eb[n] - bm);
    red[t] = sum; __syncthreads();
    for (int s = 128; s > 0; s >>= 1) { if (t < s) red[t] += red[t + s]; __syncthreads(); }
    if (t == 0) { mx[b] = bm; sm[b] = red[0]; }
}

// ---- Kernel 5: stream V, accumulate weights^T @ V partials; emit weights ----
// grid (nc=16, vc=4, b=32), 256 threads. Wave ww handles 16 rows, lane holds
// 4 columns (float4 / global_load_b128). Deterministic LDS tree combine.
__global__ void k_wv(const float* __restrict__ V, const float* __restrict__ e,
                     const float* __restrict__ mx, const float* __restrict__ sm,
                     float* __restrict__ part, float* __restrict__ outW) {
    int nc = blockIdx.x, vc = blockIdx.y, b = blockIdx.z;
    int t = threadIdx.x, lane = t & 31, ww = t >> 5;
    int n0 = nc * 128 + ww * 16;
    int v0 = vc * 128;
    float bm  = mx[b];
    float inv = 1.0f / sm[b];
    const float* eb = e + (size_t)b * NKK;
    const float* Vb = V + (size_t)b * NKK * DVV + v0 + lane * 4;
    float4 acc = make_float4(0.f, 0.f, 0.f, 0.f);
#pragma unroll 4
    for (int i = 0; i < 16; ++i) {
        int n = n0 + i;
        float wt = __expf(eb[n] - bm) * inv;
        float4 v4 = *(const float4*)(Vb + (size_t)n * DVV);
        acc.x += wt * v4.x; acc.y += wt * v4.y;
        acc.z += wt * v4.z; acc.w += wt * v4.w;
        if (vc == 0 && lane == 0) outW[(size_t)b * NKK + n] = wt;
    }
    __shared__ float sred[8 * 128];
    float* my = &sred[ww * 128 + lane * 4];
    my[0] = acc.x; my[1] = acc.y; my[2] = acc.z; my[3] = acc.w;
    __syncthreads();
    if (t < 128) {
        float s = 0.f;
#pragma unroll
        for (int w2 = 0; w2 < 8; ++w2) s += sred[w2 * 128 + t];  // fixed order
        part[((size_t)(b * 16 + nc) * 4 + vc) * 128 + t] = s;
    }
}

// ---- Kernel 6: deterministic final reduce over the 16 n-chunks ----
__global__ void k_fin(const float* __restrict__ part, float* __restrict__ out) {
    int b = blockIdx.x, t = threadIdx.x;
    for (int v = t; v < DVV; v += 256) {
        int vc = v >> 7, col = v & 127;
        float s = 0.f;
#pragma unroll
        for (int nc = 0; nc < 16; ++nc)
            s += part[((size_t)(b * 16 + nc) * 4 + vc) * 128 + col];
        out[(size_t)b * DVV + v] = s;
    }
}

extern "C" void kernel_launch(void* const* d_in, const int* in_sizes, int n_in,
                              void* d_out, int out_size, void* d_ws, size_t ws_size,
                              hipStream_t stream) {
    const float* Q  = (const float*)d_in[0];
    const float* K  = (const float*)d_in[1];
    const float* V  = (const float*)d_in[2];
    const float* Wk = (const float*)d_in[3];
    const float* bk = (const float*)d_in[4];
    const float* Wq = (const float*)d_in[5];
    const float* bq = (const float*)d_in[6];
    const float* Wh = (const float*)d_in[7];
    const float* bh = (const float*)d_in[8];

    float* out  = (float*)d_out;            // [32,512] output
    float* outW = out + BQ * DVV;           // [32,2048] weights
    float* f = (float*)d_ws;                // needs ~1.26 MB of scratch

    k_wu    <<<128, 256, 0, stream>>>(Wk, Wq, Wh, f + WS_W, f + WS_U);
    k_ctx   <<<BQ,   32, 0, stream>>>(Q, f + WS_U, bk, bq, Wh, bh, f + WS_C);
    k_energy<<<512, 256, 0, stream>>>(K, f + WS_W, f + WS_C, f + WS_E);
    k_stats <<<BQ,  256, 0, stream>>>(f + WS_E, f + WS_MAX, f + WS_SUM);
    k_wv    <<<dim3(16, 4, BQ), 256, 0, stream>>>(V, f + WS_E, f + WS_MAX,
                                                  f + WS_SUM, f + WS_PART, outW);
    k_fin   <<<BQ,  256, 0, stream>>>(f + WS_PART, out);
}